// DiffeomorphicTransform_19069654794932
// MI455X (gfx1250) — compile-verified
//
#include <hip/hip_runtime.h>

// DiffeomorphicTransform (scaling & squaring, 7 steps) for MI455X / gfx1250.
//
// flow0 = velocity * 2^-7
// repeat 7x: flow <- flow + trilinear_sample(flow, grid + flow*range_flow)
//
// [3,160,192,144] fp32, N=4,423,680 voxels. Working set (grid 53MB + two
// flow buffers 106MB) fits in the 192MB global L2, so all gathers after the
// first touch are L2 hits; default RT temporal hints keep them resident.
// WMMA is inapplicable (data-dependent gathers, no shared GEMM operand);
// this is a bandwidth/latency-bound kernel. gfx1250-specific path used:
// global_prefetch_b8 lookahead on the streaming interleaved grid.

namespace {

constexpr int TIME_STEP = 7;
constexpr int D  = 160;
constexpr int H  = 192;
constexpr int W  = 144;
constexpr int HW = H * W;          // 27648
constexpr int N  = D * HW;         // 4,423,680 voxels (divisible by 256)
constexpr int BLK = 256;           // 8 wave32 waves per block

__global__ __launch_bounds__(BLK)
void diffeo_compose(const float* __restrict__ src,    // [3,D,H,W] (channel-major)
                    const float* __restrict__ grid,   // [D,H,W,3] (interleaved x,y,z)
                    const float* __restrict__ rf_ptr, // device scalar range_flow
                    float* __restrict__ dst,          // [3,D,H,W]
                    float scale)                      // 2^-7 on first pass, else 1.0
{
    const int i = blockIdx.x * BLK + threadIdx.x;
    if (i >= N) return;

    const float rf = rf_ptr[0];

    // Self flow (coalesced), with first-pass scaling fused in.
    float fx = src[i]         * scale;
    float fy = src[i + N]     * scale;
    float fz = src[i + 2 * N] * scale;

    const int gi = 3 * i;
    // Streaming lookahead prefetch of the interleaved grid
    // (lowered to global_prefetch_b8 on gfx1250).
    __builtin_prefetch(grid + gi + 3 * BLK, 0, 3);

    // Sample position in voxel coordinates (normalize/unnormalize cancels
    // exactly for align_corners=True).
    const float px = grid[gi + 0] + fx * rf;
    const float py = grid[gi + 1] + fy * rf;
    const float pz = grid[gi + 2] + fz * rf;

    const float flx = floorf(px), fly = floorf(py), flz = floorf(pz);
    const float ax = px - flx, ay = py - fly, az = pz - flz;
    const int ix0 = (int)flx, iy0 = (int)fly, iz0 = (int)flz;
    const int ix1 = ix0 + 1,  iy1 = iy0 + 1,  iz1 = iz0 + 1;

    // Per-axis weights with the zeros-padding validity mask folded in
    // (corner mask == product of axis masks). Fold the sampled-value scale
    // into the z weights so gathered values need no extra multiply.
    const float wx0 = ((unsigned)ix0 < (unsigned)W) ? (1.0f - ax) : 0.0f;
    const float wx1 = ((unsigned)ix1 < (unsigned)W) ? ax          : 0.0f;
    const float wy0 = ((unsigned)iy0 < (unsigned)H) ? (1.0f - ay) : 0.0f;
    const float wy1 = ((unsigned)iy1 < (unsigned)H) ? ay          : 0.0f;
    const float wz0 = (((unsigned)iz0 < (unsigned)D) ? (1.0f - az) : 0.0f) * scale;
    const float wz1 = (((unsigned)iz1 < (unsigned)D) ? az          : 0.0f) * scale;

    // Clamped indices (loads are always in-bounds; masks zero the weight).
    const int cx0 = min(max(ix0, 0), W - 1), cx1 = min(max(ix1, 0), W - 1);
    const int cy0 = min(max(iy0, 0), H - 1), cy1 = min(max(iy1, 0), H - 1);
    const int cz0 = min(max(iz0, 0), D - 1), cz1 = min(max(iz1, 0), D - 1);

    const int b00 = cz0 * HW + cy0 * W;
    const int b01 = cz0 * HW + cy1 * W;
    const int b10 = cz1 * HW + cy0 * W;
    const int b11 = cz1 * HW + cy1 * W;

    const float wz0y0 = wz0 * wy0, wz0y1 = wz0 * wy1;
    const float wz1y0 = wz1 * wy0, wz1y1 = wz1 * wy1;

    float sx = 0.0f, sy = 0.0f, sz = 0.0f;

#define DIFFEO_CORNER(base, wzy, wx, cx)              \
    {                                                 \
        const int   o = (base) + (cx);                \
        const float w = (wzy) * (wx);                 \
        sx = fmaf(w, src[o],         sx);             \
        sy = fmaf(w, src[o + N],     sy);             \
        sz = fmaf(w, src[o + 2 * N], sz);             \
    }

    DIFFEO_CORNER(b00, wz0y0, wx0, cx0)
    DIFFEO_CORNER(b00, wz0y0, wx1, cx1)
    DIFFEO_CORNER(b01, wz0y1, wx0, cx0)
    DIFFEO_CORNER(b01, wz0y1, wx1, cx1)
    DIFFEO_CORNER(b10, wz1y0, wx0, cx0)
    DIFFEO_CORNER(b10, wz1y0, wx1, cx1)
    DIFFEO_CORNER(b11, wz1y1, wx0, cx0)
    DIFFEO_CORNER(b11, wz1y1, wx1, cx1)
#undef DIFFEO_CORNER

    // flow <- flow + sampled(flow)   (coalesced stores, RT -> stays in L2
    // for the next iteration's gathers)
    dst[i]         = fx + sx;
    dst[i + N]     = fy + sy;
    dst[i + 2 * N] = fz + sz;
}

} // namespace

extern "C" void kernel_launch(void* const* d_in, const int* in_sizes, int n_in,
                              void* d_out, int out_size, void* d_ws, size_t ws_size,
                              hipStream_t stream) {
    (void)in_sizes; (void)n_in; (void)out_size; (void)ws_size;

    const float* velocity = (const float*)d_in[0];   // [1,3,160,192,144] fp32
    const float* grid     = (const float*)d_in[1];   // [1,160,192,144,3] fp32
    const float* rf       = (const float*)d_in[2];   // device scalar

    float* out = (float*)d_out;                      // ping buffer (final result)
    float* ws  = (float*)d_ws;                       // pong buffer (needs 3*N*4 = 53MB)

    const int nblk = (N + BLK - 1) / BLK;            // 17280 blocks, no tail

    // 7 composition passes. Pass 0 reads velocity with the 2^-7 scaling fused
    // into the weights; subsequent passes ping-pong d_out <-> d_ws. Odd pass
    // count lands the final flow in d_out. Inputs are never written.
    const float* src = velocity;
    for (int t = 0; t < TIME_STEP; ++t) {
        float* dst = ((t & 1) == 0) ? out : ws;
        const float scale = (t == 0) ? (1.0f / 128.0f) : 1.0f;
        diffeo_compose<<<nblk, BLK, 0, stream>>>(src, grid, rf, dst, scale);
        src = dst;
    }
}